// StrengthChangeDetectionModule_7043746365858
// MI455X (gfx1250) — compile-verified
//
#include <hip/hip_runtime.h>

typedef float v2f __attribute__((ext_vector_type(2)));
typedef float v8f __attribute__((ext_vector_type(8)));

#define B_DIM 128
#define S_DIM 524288
#define T_DIM 256
#define L_DIM 2048          // S_DIM / T_DIM, elements per segment
#define CHUNKS 16           // L_DIM / 128: 128 elements (4 per lane) per chunk
#define GAPS_F 2047.0f      // L_DIM - 1

// One wave32 per segment. Streams the segment with b128 loads; each lane also
// fetches its boundary element with a coalesced b32 from the same 512B window
// (no extra HBM traffic, no shuffle, no EXEC divergence). The 2047 |diff|
// values are reduced with V_WMMA_F32_16X16X4_F32 against an all-ones B matrix:
// D row-sums accumulate EVERY element of A, so the A lane layout is irrelevant
// to the total and the reduction stays exact fp32.
__global__ __launch_bounds__(256) void
segment_absdiff_mean_kernel(const float* __restrict__ x, float* __restrict__ out) {
    const int lane    = threadIdx.x & 31;
    const int wave    = threadIdx.x >> 5;
    const int seg_id  = blockIdx.x * 8 + wave;          // 0 .. 32767
    const int b       = seg_id >> 8;                    // / T_DIM
    const int t       = seg_id & (T_DIM - 1);
    const float* __restrict__ seg = x + (size_t)b * S_DIM + (size_t)t * L_DIM;

    v8f acc = {};                        // C/D accumulator (16x16 f32, 8 VGPRs)
    const v2f ones = {1.0f, 1.0f};       // B matrix: all ones

    #pragma unroll
    for (int c = 0; c < CHUNKS; ++c) {
        const int base = c * 128 + 4 * lane;
        // 512B contiguous per wave, 16B aligned per lane -> global_load_b128
        const float4 v = *(const float4*)(seg + base);

        // Boundary element x[base+4]: in-bounds for every (c,lane) except
        // c==15, lane==31 (local gap index 2047 doesn't exist). Branch-free:
        // clamp the address into the segment and zero that one diff.
        const int  nidx  = base + 4;
        const bool valid = (nidx < L_DIM);
        const float nx   = seg[valid ? nidx : (L_DIM - 1)];

        v2f a0, a1;
        a0.x = fabsf(v.y - v.x);
        a0.y = fabsf(v.z - v.y);
        a1.x = fabsf(v.w - v.z);
        a1.y = valid ? fabsf(nx - v.w) : 0.0f;

        // D = A x ones + C : adds all 64 values of A into the row sums.
        acc = __builtin_amdgcn_wmma_f32_16x16x4_f32(
                  false, a0, false, ones, (short)0, acc, false, false);
        acc = __builtin_amdgcn_wmma_f32_16x16x4_f32(
                  false, a1, false, ones, (short)0, acc, false, false);
    }

    // acc VGPR j holds row M=j (lanes 0-15) / M=j+8 (lanes 16-31), replicated
    // across the 16 lanes of each half. Sum the 8 VGPRs, then combine the two
    // half-wave row groups with one cross-half shuffle.
    float s = acc[0] + acc[1] + acc[2] + acc[3] +
              acc[4] + acc[5] + acc[6] + acc[7];
    s += __shfl_xor(s, 16, 32);

    if (lane == 0) {
        out[seg_id] = s / GAPS_F;
    }
}

extern "C" void kernel_launch(void* const* d_in, const int* in_sizes, int n_in,
                              void* d_out, int out_size, void* d_ws, size_t ws_size,
                              hipStream_t stream) {
    const float* x = (const float*)d_in[0];
    float* out     = (float*)d_out;
    (void)in_sizes; (void)n_in; (void)d_ws; (void)ws_size; (void)out_size;

    const int total_segments = B_DIM * T_DIM;        // 32768 waves
    dim3 grid(total_segments / 8);                   // 8 waves per block
    dim3 block(256);
    hipLaunchKernelGGL(segment_absdiff_mean_kernel, grid, block, 0, stream, x, out);
}